// MOERouter_58901181498108
// MI455X (gfx1250) — compile-verified
//
#include <hip/hip_runtime.h>
#include <hip/hip_bf16.h>

// MI455X / gfx1250 MoE router.
// Phase 1: router logits via V_WMMA_F32_16X16X4_F32 (memory-bound GEMM, f32).
// Phase 2: softmax + top-2 + normalize + one-hot mask, one thread per token.

typedef __attribute__((ext_vector_type(2))) float v2f;
typedef __attribute__((ext_vector_type(8))) float v8f;

#define NUM_TOKENS 16384
#define HIDDEN     2048
#define NUM_EXPERTS 16

// ---------------------------------------------------------------------------
// Kernel 1: logits[T,16] = x[T,2048] @ W[16,2048]^T + b
// One wave32 per 16-token tile. K-loop of 512 x (16x16x4) f32 WMMAs.
// A-frag layout (ISA 16x4 f32 A): lane m<16 holds K={0,1}, lanes 16..31 hold
// K={2,3} for the same rows -> a float2 load at (row, k0 + 2*(lane>>4)).
// B-frag is W^T 4x16 with lane = expert column; same half-wave K split.
// ---------------------------------------------------------------------------
__global__ __launch_bounds__(256) void moe_logits_wmma(
    const float* __restrict__ x,      // [T, H]
    const float* __restrict__ W,      // [E, H]
    const float* __restrict__ b,      // [E]
    float* __restrict__ logits)       // [T, E]
{
    const int lane  = threadIdx.x & 31;
    const int wave  = threadIdx.x >> 5;
    const int tile  = blockIdx.x * 8 + wave;   // which 16-token tile
    const int m     = lane & 15;               // token-in-tile (A) / expert (B)
    const int khalf = (lane >> 4) << 1;        // 0 for lanes 0-15, 2 for 16-31

    const float* __restrict__ xrow = x + (size_t)(tile * 16 + m) * HIDDEN + khalf;
    const float* __restrict__ wrow = W + (size_t)m * HIDDEN + khalf;

    v8f c = {0.f, 0.f, 0.f, 0.f, 0.f, 0.f, 0.f, 0.f};

    for (int k0 = 0; k0 < HIDDEN; k0 += 32) {
        // stream-ahead prefetch on the big x stream (global_prefetch_b8)
        __builtin_prefetch(xrow + k0 + 512, 0, 1);
#pragma unroll
        for (int kk = 0; kk < 32; kk += 4) {
            v2f a  = *(const v2f*)(xrow + k0 + kk);
            v2f bb = *(const v2f*)(wrow + k0 + kk);
            // D = A(16x4) x B(4x16) + C, f32
            c = __builtin_amdgcn_wmma_f32_16x16x4_f32(
                    /*neg_a=*/false, a, /*neg_b=*/false, bb,
                    /*c_mod=*/(short)0, c, /*reuse_a=*/false, /*reuse_b=*/false);
        }
    }

    // C/D layout: VGPR r holds token M=r (lanes 0-15) or M=8+r (lanes 16-31),
    // expert N = lane & 15.
    const int   e       = m;
    const float bias    = b[e];
    const int   tokBase = tile * 16 + ((lane >> 4) << 3);
#pragma unroll
    for (int r = 0; r < 8; ++r) {
        logits[(size_t)(tokBase + r) * NUM_EXPERTS + e] = c[r] + bias;
    }
}

// ---------------------------------------------------------------------------
// Kernel 2: per-token softmax + top-2 (jax tie-break: lowest index first),
// normalized weights, indices (as float), one-hot mask [E, K, T].
// ---------------------------------------------------------------------------
__global__ __launch_bounds__(256) void moe_router_topk(
    const float* __restrict__ logits,  // [T, E]
    float* __restrict__ wout,          // [T, 2]
    float* __restrict__ iout,          // [T, 2] (indices as float)
    float* __restrict__ mout)          // [E, 2, T]
{
    const int t = blockIdx.x * blockDim.x + threadIdx.x;
    if (t >= NUM_TOKENS) return;

    float l[NUM_EXPERTS];
    const float4* lp = (const float4*)(logits + (size_t)t * NUM_EXPERTS);
#pragma unroll
    for (int i = 0; i < 4; ++i) {
        float4 v = lp[i];
        l[4 * i + 0] = v.x; l[4 * i + 1] = v.y;
        l[4 * i + 2] = v.z; l[4 * i + 3] = v.w;
    }

    // softmax with max subtraction
    float mx = l[0];
#pragma unroll
    for (int i = 1; i < NUM_EXPERTS; ++i) mx = fmaxf(mx, l[i]);
    float p[NUM_EXPERTS];
    float s = 0.f;
#pragma unroll
    for (int i = 0; i < NUM_EXPERTS; ++i) { p[i] = __expf(l[i] - mx); s += p[i]; }
    const float inv = 1.f / s;
#pragma unroll
    for (int i = 0; i < NUM_EXPERTS; ++i) p[i] *= inv;

    // top-2, stable (strict > keeps first occurrence on ties, matching jax)
    int i0 = 0; float p0 = p[0];
#pragma unroll
    for (int i = 1; i < NUM_EXPERTS; ++i) {
        if (p[i] > p0) { p0 = p[i]; i0 = i; }
    }
    int i1 = -1; float p1 = -1.0e30f;
#pragma unroll
    for (int i = 0; i < NUM_EXPERTS; ++i) {
        if (i != i0 && p[i] > p1) { p1 = p[i]; i1 = i; }
    }

    const float invsum = 1.f / (p0 + p1);
    wout[2 * t + 0] = p0 * invsum;
    wout[2 * t + 1] = p1 * invsum;
    iout[2 * t + 0] = (float)i0;
    iout[2 * t + 1] = (float)i1;

    // expert_mask [E, 2, T]; each store is coalesced across the wave in t.
#pragma unroll
    for (int e = 0; e < NUM_EXPERTS; ++e) {
        mout[((size_t)e * 2 + 0) * NUM_TOKENS + t] = (i0 == e) ? 1.f : 0.f;
        mout[((size_t)e * 2 + 1) * NUM_TOKENS + t] = (i1 == e) ? 1.f : 0.f;
    }
}

extern "C" void kernel_launch(void* const* d_in, const int* in_sizes, int n_in,
                              void* d_out, int out_size, void* d_ws, size_t ws_size,
                              hipStream_t stream) {
    const float* x = (const float*)d_in[0];   // [16384, 2048]
    const float* W = (const float*)d_in[1];   // [16, 2048]
    const float* b = (const float*)d_in[2];   // [16]

    float* out    = (float*)d_out;
    float* logits = out;                                   // 16384*16
    float* wout   = logits + (size_t)NUM_TOKENS * NUM_EXPERTS;  // 16384*2
    float* iout   = wout   + (size_t)NUM_TOKENS * 2;            // 16384*2
    float* mout   = iout   + (size_t)NUM_TOKENS * 2;            // 16*2*16384

    // 8 waves/block * 16 tokens/wave = 128 tokens/block -> 128 blocks
    moe_logits_wmma<<<NUM_TOKENS / 128, 256, 0, stream>>>(x, W, b, logits);
    // one thread per token
    moe_router_topk<<<NUM_TOKENS / 256, 256, 0, stream>>>(logits, wout, iout, mout);
}